// Attention_70411693851344
// MI455X (gfx1250) — compile-verified
//
#include <hip/hip_runtime.h>

typedef __attribute__((ext_vector_type(2))) float v2f;
typedef __attribute__((ext_vector_type(4))) float v4f;
typedef __attribute__((ext_vector_type(8))) float v8f;

#define D_MODEL 2048
#define NH 16
#define HEAD_DIM 128
#define KV_LEN 4096
#define BATCH 16

// ---------------------------------------------------------------------------
// GEMM tile: out[m][n] = sum_k A[m][k] * W[n][k]   (A: [16 x 2048] row-major,
// W: [2048 x 2048] row-major, i.e. out = A @ W^T). One wave32 computes one
// 16x16 output tile via V_WMMA_F32_16X16X4_F32, K-loop of 512 steps.
// Weights are loaded with default (RT) temporal hint so the 192 MB L2 can
// retain them across graph replays; the K/V streams below are NT.
// ---------------------------------------------------------------------------
__device__ __forceinline__ void gemm_tile_xWt(const float* __restrict__ A,
                                              const float* __restrict__ W,
                                              float* __restrict__ out,
                                              int tile_n) {
  const int lane  = threadIdx.x & 31;
  const int m     = lane & 15;
  const int khalf = (lane >> 4) << 1;          // 0 or 2
  const int n     = tile_n * 16 + m;           // B column owned by this lane
  const float* arow = A + m * D_MODEL + khalf;
  const float* wrow = W + n * D_MODEL + khalf;

  v8f acc = {};
#pragma unroll 4
  for (int kk = 0; kk < D_MODEL; kk += 4) {
    v2f a = *(const v2f*)(arow + kk);
    v2f b = *(const v2f*)(wrow + kk);
    // 8 args: (neg_a, A, neg_b, B, c_mod, C, reuse_a, reuse_b)
    acc = __builtin_amdgcn_wmma_f32_16x16x4_f32(false, a, false, b,
                                                (short)0, acc, false, false);
  }

  const int mbase = (lane >> 4) * 8;           // lanes 16-31 hold M=8..15
  const int col   = tile_n * 16 + (lane & 15);
#pragma unroll
  for (int r = 0; r < 8; ++r) {
    out[(mbase + r) * D_MODEL + col] = acc[r];
  }
}

// ws layout (floats): [0) q[16][2048] | [32768) k_new | [65536) v_new | [98304) attn_out
__global__ void qkv_gemm_kernel(const float* __restrict__ x,
                                const float* __restrict__ Wq,
                                const float* __restrict__ Wk,
                                const float* __restrict__ Wv,
                                float* __restrict__ ws) {
  const int tile  = blockIdx.x;        // 0..383
  const int which = tile >> 7;         // 0=q, 1=k, 2=v
  const int tn    = tile & 127;
  const float* W = (which == 0) ? Wq : (which == 1) ? Wk : Wv;
  float* out = ws + which * (BATCH * D_MODEL);
  gemm_tile_xWt(x, W, out, tn);
}

__global__ void out_proj_kernel(const float* __restrict__ attn,
                                const float* __restrict__ Wo,
                                float* __restrict__ out) {
  gemm_tile_xWt(attn, Wo, out, blockIdx.x);
}

// ---------------------------------------------------------------------------
// Attention: one block of 256 threads (8 wave32) per (b, h). K/V rows are
// streamed once with non-temporal b128 loads. Pass 1 processes 4 rows per
// wave iteration: lane l owns row (l>>3) of the group and the 16-float chunk
// (l&7)*16, so the score reduction is a 3-step butterfly over 8 lanes instead
// of a 5-step one per single row (~2x fewer issue slots per byte streamed).
// The cache row at `position` is substituted on the fly with the freshly
// projected k/v (inputs are immutable). Two-pass softmax staged in LDS.
// ---------------------------------------------------------------------------
__global__ void attn_kernel(const float* __restrict__ k_cache,
                            const float* __restrict__ v_cache,
                            const float* __restrict__ ws,
                            const int* __restrict__ position,
                            float* __restrict__ attn_out) {
  __shared__ float s_scores[KV_LEN];           // 16 KB
  __shared__ float s_red[8];
  __shared__ float s_part[8 * HEAD_DIM];       // 4 KB

  const int b    = blockIdx.x >> 4;
  const int h    = blockIdx.x & 15;
  const int tid  = threadIdx.x;
  const int lane = tid & 31;
  const int wave = tid >> 5;
  const int pos  = position[0];

  const float* q    = ws +                       (size_t)b * D_MODEL + h * HEAD_DIM;
  const float* knew = ws + 1 * BATCH * D_MODEL + (size_t)b * D_MODEL + h * HEAD_DIM;
  const float* vnew = ws + 2 * BATCH * D_MODEL + (size_t)b * D_MODEL + h * HEAD_DIM;
  const size_t base = ((size_t)b * NH + h) * (size_t)KV_LEN * HEAD_DIM;
  const float* K = k_cache + base;
  const float* V = v_cache + base;

  const float scale = 0.08838834764831845f;    // 1/sqrt(128)

  const int jbeg = wave * (KV_LEN / 8);
  const int jend = jbeg + (KV_LEN / 8);

  // ---- pass 1: scores = (q . K_j) * scale, 4 rows per wave iteration ----
  const int sub = lane >> 3;                   // row within 4-row group
  const int c8  = lane & 7;                    // 16-float chunk within row
  const float* qp = q + c8 * 16;
  const v4f q0 = *(const v4f*)(qp + 0);
  const v4f q1 = *(const v4f*)(qp + 4);
  const v4f q2 = *(const v4f*)(qp + 8);
  const v4f q3 = *(const v4f*)(qp + 12);

  float wmax = -3.0e38f;
  for (int jb = jbeg; jb < jend; jb += 4) {
    const int j = jb + sub;
    const float* krow = (j == pos) ? knew : (K + (size_t)j * HEAD_DIM);
    const float* kp = krow + c8 * 16;
    const v4f k0 = __builtin_nontemporal_load((const v4f*)(kp + 0));
    const v4f k1 = __builtin_nontemporal_load((const v4f*)(kp + 4));
    const v4f k2 = __builtin_nontemporal_load((const v4f*)(kp + 8));
    const v4f k3 = __builtin_nontemporal_load((const v4f*)(kp + 12));
    float p = q0.x * k0.x + q0.y * k0.y + q0.z * k0.z + q0.w * k0.w;
    p += q1.x * k1.x + q1.y * k1.y + q1.z * k1.z + q1.w * k1.w;
    p += q2.x * k2.x + q2.y * k2.y + q2.z * k2.z + q2.w * k2.w;
    p += q3.x * k3.x + q3.y * k3.y + q3.z * k3.z + q3.w * k3.w;
    // reduce across the 8 lanes sharing this row
    p += __shfl_xor(p, 1, 32);
    p += __shfl_xor(p, 2, 32);
    p += __shfl_xor(p, 4, 32);
    p *= scale;
    if (c8 == 0) s_scores[j] = p;
    wmax = fmaxf(wmax, p);                     // per-lane: max over this lane's rows
  }
  // one-time full-wave max (each 8-lane group tracked different rows)
#pragma unroll
  for (int off = 16; off >= 1; off >>= 1) wmax = fmaxf(wmax, __shfl_xor(wmax, off, 32));
  if (lane == 0) s_red[wave] = wmax;
  __syncthreads();
  float gmax = s_red[0];
#pragma unroll
  for (int w = 1; w < 8; ++w) gmax = fmaxf(gmax, s_red[w]);
  __syncthreads();                             // protect s_red before reuse

  // ---- pass 2: exponentiate in place, accumulate denominator ----
  float lsum = 0.f;
  for (int j = tid; j < KV_LEN; j += 256) {
    const float e = __expf(s_scores[j] - gmax);
    s_scores[j] = e;
    lsum += e;
  }
#pragma unroll
  for (int off = 16; off >= 1; off >>= 1) lsum += __shfl_xor(lsum, off, 32);
  if (lane == 0) s_red[wave] = lsum;
  __syncthreads();
  float gsum = 0.f;
#pragma unroll
  for (int w = 0; w < 8; ++w) gsum += s_red[w];
  const float inv = 1.0f / gsum;

  // ---- pass 3: out = sum_j p_j * V_j (lane l owns d = 4l..4l+3) ----
  v4f acc = {};
  for (int j = jbeg; j < jend; ++j) {
    const float* vrow = (j == pos) ? vnew : (V + (size_t)j * HEAD_DIM);
    const v4f vc = __builtin_nontemporal_load((const v4f*)(vrow + 4 * lane));
    const float p = s_scores[j];
    acc.x += p * vc.x; acc.y += p * vc.y; acc.z += p * vc.z; acc.w += p * vc.w;
  }
  float* sp = s_part + wave * HEAD_DIM + 4 * lane;
  sp[0] = acc.x; sp[1] = acc.y; sp[2] = acc.z; sp[3] = acc.w;
  __syncthreads();
  if (tid < HEAD_DIM) {
    float s = 0.f;
#pragma unroll
    for (int w = 0; w < 8; ++w) s += s_part[w * HEAD_DIM + tid];
    attn_out[(size_t)b * D_MODEL + h * HEAD_DIM + tid] = s * inv;
  }
}

// ---------------------------------------------------------------------------
extern "C" void kernel_launch(void* const* d_in, const int* in_sizes, int n_in,
                              void* d_out, int out_size, void* d_ws, size_t ws_size,
                              hipStream_t stream) {
  const float* x   = (const float*)d_in[0];
  const float* Wq  = (const float*)d_in[1];
  const float* Wk  = (const float*)d_in[2];
  const float* Wv  = (const float*)d_in[3];
  const float* Wo  = (const float*)d_in[4];
  const float* kc  = (const float*)d_in[5];
  const float* vc  = (const float*)d_in[6];
  const int*   pos = (const int*)d_in[7];
  float* out = (float*)d_out;
  float* ws  = (float*)d_ws;
  float* attn = ws + 3 * BATCH * D_MODEL;

  // 3 weight matrices x 128 N-tiles, one wave per tile
  qkv_gemm_kernel<<<dim3(384), dim3(32), 0, stream>>>(x, Wq, Wk, Wv, ws);
  // one block per (b, h)
  attn_kernel<<<dim3(BATCH * NH), dim3(256), 0, stream>>>(kc, vc, ws, pos, attn);
  // 128 N-tiles for the output projection
  out_proj_kernel<<<dim3(128), dim3(32), 0, stream>>>(attn, Wo, out);
}